// SelectiveSSM_17068200035056
// MI455X (gfx1250) — compile-verified
//
#include <hip/hip_runtime.h>

namespace {
constexpr int kDConv  = 4;
constexpr int kDInner = 64;
constexpr int kB = 8;
constexpr int kL = 8192;
constexpr int kT = kB * kL;            // 65536 tokens
constexpr int kChunk  = 128;
constexpr int kNChunk = kL / kChunk;   // 64
}

typedef float v2f __attribute__((ext_vector_type(2)));
typedef float v8f __attribute__((ext_vector_type(8)));

__device__ __forceinline__ float silu_f(float x)     { return x / (1.0f + expf(-x)); }
__device__ __forceinline__ float softplus_f(float x) { return x > 20.0f ? x : log1pf(expf(x)); }

// One wave computes one 16x16 fp32 C-tile of  A[MxK] * W^T  (W is N-major, ldw=K)
// using V_WMMA_F32_16X16X4_F32.  A-frag: lane holds row m=lane%16, K pair at
// k0 = 4*step + (lane>=16 ? 2 : 0).  B-frag mirrored with n=lane%16.
template <int K>
__device__ __forceinline__ v8f wave_gemm_f32(const float* __restrict__ A, int lda,
                                             const float* __restrict__ W, int ldw,
                                             int m0, int n0, unsigned lane) {
  const unsigned mr = lane & 15u;
  const unsigned hi = lane >> 4;                 // K-half select (0/1)
  v8f acc = {0.f, 0.f, 0.f, 0.f, 0.f, 0.f, 0.f, 0.f};
  const float* arow = A + (size_t)(m0 + (int)mr) * lda;
  const float* wrow = W + (size_t)(n0 + (int)mr) * ldw;
  #pragma unroll
  for (int k = 0; k < K; k += 4) {
    const int k0 = k + (int)(hi * 2u);
    v2f a; a.x = arow[k0]; a.y = arow[k0 + 1];
    v2f b; b.x = wrow[k0]; b.y = wrow[k0 + 1];
    acc = __builtin_amdgcn_wmma_f32_16x16x4_f32(false, a, false, b, (short)0, acc,
                                                false, false);
  }
  return acc;
}

// ---- Phase 1: xz = x @ W_in^T + b_in ; split -> xi, silu(z) ----------------
__global__ void __launch_bounds__(256)
k_gemm_in(const float* __restrict__ x, const float* __restrict__ W_in,
          const float* __restrict__ b_in, float* __restrict__ xi,
          float* __restrict__ zs) {
  const unsigned tid = threadIdx.x, lane = tid & 31u;
  const int mtile = blockIdx.x;
  const int ntile = (int)(tid >> 5);                       // 8 waves -> N=128
  v8f c = wave_gemm_f32<32>(x, 32, W_in, 32, mtile * 16, ntile * 16, lane);
  const int n = ntile * 16 + (int)(lane & 15u);
  const float bias = b_in[n];
  const int mbase = mtile * 16 + (int)((lane >> 4) * 8u);
  #pragma unroll
  for (int r = 0; r < 8; ++r) {
    const int m = mbase + r;
    const float v = c[r] + bias;
    if (n < kDInner) xi[(size_t)m * kDInner + n] = v;
    else             zs[(size_t)m * kDInner + (n - kDInner)] = silu_f(v);
  }
}

// ---- Phase 2: depthwise causal conv (k=4) + SiLU -> ut ---------------------
__global__ void __launch_bounds__(256)
k_conv(const float* __restrict__ xi, const float* __restrict__ W_conv,
       const float* __restrict__ b_conv, float* __restrict__ ut) {
  const int gid = blockIdx.x * 256 + (int)threadIdx.x;
  const int d   = gid & (kDInner - 1);
  const int tok = gid >> 6;
  const int l   = tok & (kL - 1);
  float acc = b_conv[d];
  #pragma unroll
  for (int k = 0; k < kDConv; ++k) {
    const int li = l - (kDConv - 1) + k;
    if (li >= 0)
      acc = fmaf(xi[(size_t)(tok - (kDConv - 1) + k) * kDInner + d],
                 W_conv[d * kDConv + k], acc);
  }
  ut[(size_t)tok * kDInner + d] = silu_f(acc);
}

// ---- Phase 3a: x_dbl = ut @ W_xp^T + b_xp ; split -> dpre, bt, ct ----------
__global__ void __launch_bounds__(256)
k_gemm_xp(const float* __restrict__ ut, const float* __restrict__ W_xp,
          const float* __restrict__ b_xp, float* __restrict__ dpre,
          float* __restrict__ bt, float* __restrict__ ct) {
  const unsigned tid = threadIdx.x, lane = tid & 31u;
  const int tile  = blockIdx.x * 8 + (int)(tid >> 5);
  const int mtile = tile / 12, ntile = tile % 12;          // N = 192
  v8f c = wave_gemm_f32<64>(ut, 64, W_xp, 64, mtile * 16, ntile * 16, lane);
  const int n = ntile * 16 + (int)(lane & 15u);
  const float bias = b_xp[n];
  const int mbase = mtile * 16 + (int)((lane >> 4) * 8u);
  #pragma unroll
  for (int r = 0; r < 8; ++r) {
    const int m = mbase + r;
    const float v = c[r] + bias;
    if (n < 64)       dpre[(size_t)m * 64 + n] = v;
    else if (n < 128) bt[(size_t)m * 64 + (n - 64)] = v;
    else              ct[(size_t)m * 64 + (n - 128)] = v;
  }
}

// ---- Phase 3b: dt = softplus(dpre @ W_dt^T + b_dt) -------------------------
__global__ void __launch_bounds__(256)
k_gemm_dt(const float* __restrict__ dpre, const float* __restrict__ W_dt,
          const float* __restrict__ b_dt, float* __restrict__ dt) {
  const unsigned tid = threadIdx.x, lane = tid & 31u;
  const int tile  = blockIdx.x * 8 + (int)(tid >> 5);
  const int mtile = tile / 4, ntile = tile % 4;            // N = 64
  v8f c = wave_gemm_f32<64>(dpre, 64, W_dt, 64, mtile * 16, ntile * 16, lane);
  const int n = ntile * 16 + (int)(lane & 15u);
  const float bias = b_dt[n];
  const int mbase = mtile * 16 + (int)((lane >> 4) * 8u);
  #pragma unroll
  for (int r = 0; r < 8; ++r)
    dt[(size_t)(mbase + r) * 64 + n] = softplus_f(c[r] + bias);
}

// ---- Phase 4a: per-chunk local scan; record local h and decay products -----
// Wave per (b, d, chunk); lane holds states n = 2*lane, 2*lane+1.
__global__ void __launch_bounds__(256)
k_scanA(const float* __restrict__ dt, const float* __restrict__ bt,
        const float* __restrict__ ut, const float* __restrict__ A_log,
        float2* __restrict__ hloc, float2* __restrict__ aprod) {
  const unsigned tid = threadIdx.x, lane = tid & 31u;
  const unsigned wid = blockIdx.x * 8u + (tid >> 5);
  const unsigned ch = wid & (kNChunk - 1);
  const unsigned bd = wid >> 6;
  const unsigned d = bd & 63u, b = bd >> 6;
  const float2 al = ((const float2*)A_log)[d * 32u + lane];
  const float ax = -expf(al.x), ay = -expf(al.y);           // A[d,n]
  float hx = 0.f, hy = 0.f, px = 1.f, py = 1.f;
  const size_t tok0 = (size_t)b * kL + (size_t)ch * kChunk;
  for (int i = 0; i < kChunk; ++i) {
    const size_t off = (tok0 + i) * 64 + d;
    const float dtv = dt[off], btv = bt[off], utv = ut[off];
    const float s = dtv * btv * utv;
    const float fx = fmaf(-dtv, ax, 1.0f);
    const float fy = fmaf(-dtv, ay, 1.0f);
    hx = fmaf(hx, fx, s); hy = fmaf(hy, fy, s);
    px *= fx; py *= fy;
  }
  const size_t o = ((size_t)(ch * 8u + b) * 64u + d) * 32u + lane;
  hloc[o]  = make_float2(hx, hy);
  aprod[o] = make_float2(px, py);
}

// ---- Phase 4b: serial prefix across the 64 chunks (in place: hloc -> hstart)
__global__ void __launch_bounds__(256)
k_scanB(float* __restrict__ hloc, const float* __restrict__ aprod) {
  const int i = blockIdx.x * 256 + (int)threadIdx.x;        // (b,d,n) flat
  float h = 0.f;
  for (int ch = 0; ch < kNChunk; ++ch) {
    const size_t idx = (size_t)ch * (kB * 64 * 64) + i;
    const float pa = aprod[idx];
    const float hl = hloc[idx];
    hloc[idx] = h;                                          // chunk start state
    h = fmaf(h, pa, hl);
  }
}

// ---- Phase 4c: replay chunk from true start; y-reduction; gate; write yg ---
__global__ void __launch_bounds__(256)
k_scanC(const float* __restrict__ dt, const float* __restrict__ bt,
        float* __restrict__ ut, const float* __restrict__ ct,
        const float* __restrict__ zs, const float2* __restrict__ hstart,
        const float* __restrict__ A_log, const float* __restrict__ D_par) {
  const unsigned tid = threadIdx.x, lane = tid & 31u;
  const unsigned wid = blockIdx.x * 8u + (tid >> 5);
  const unsigned ch = wid & (kNChunk - 1);
  const unsigned bd = wid >> 6;
  const unsigned d = bd & 63u, b = bd >> 6;
  const float2 al = ((const float2*)A_log)[d * 32u + lane];
  const float ax = -expf(al.x), ay = -expf(al.y);
  const float Dd = D_par[d];
  const size_t o = ((size_t)(ch * 8u + b) * 64u + d) * 32u + lane;
  float2 h0 = hstart[o];
  float hx = h0.x, hy = h0.y;
  const size_t tok0 = (size_t)b * kL + (size_t)ch * kChunk;
  for (int i = 0; i < kChunk; ++i) {
    const size_t tok = tok0 + i;
    const size_t off = tok * 64 + d;
    const float dtv = dt[off], btv = bt[off], utv = ut[off];
    const float2 c2 = ((const float2*)(ct + tok * 64))[lane];
    const float s = dtv * btv * utv;
    hx = fmaf(hx, fmaf(-dtv, ax, 1.0f), s);
    hy = fmaf(hy, fmaf(-dtv, ay, 1.0f), s);
    float part = fmaf(hx, c2.x, hy * c2.y);
    #pragma unroll
    for (int m = 16; m >= 1; m >>= 1) part += __shfl_xor(part, m, 32);
    if (lane == 0) ut[off] = fmaf(Dd, utv, part) * zs[off]; // gated y (in place)
  }
}

// ---- Phase 5: out = yg @ W_out^T + b_out -----------------------------------
__global__ void __launch_bounds__(256)
k_gemm_out(const float* __restrict__ yg, const float* __restrict__ W_out,
           const float* __restrict__ b_out, float* __restrict__ out) {
  const unsigned tid = threadIdx.x, lane = tid & 31u;
  const int tile  = blockIdx.x * 8 + (int)(tid >> 5);
  const int mtile = tile / 2, ntile = tile % 2;             // N = 32
  v8f c = wave_gemm_f32<64>(yg, 64, W_out, 64, mtile * 16, ntile * 16, lane);
  const int n = ntile * 16 + (int)(lane & 15u);
  const float bias = b_out[n];
  const int mbase = mtile * 16 + (int)((lane >> 4) * 8u);
  #pragma unroll
  for (int r = 0; r < 8; ++r)
    out[(size_t)(mbase + r) * 32 + n] = c[r] + bias;
}

extern "C" void kernel_launch(void* const* d_in, const int* in_sizes, int n_in,
                              void* d_out, int out_size, void* d_ws, size_t ws_size,
                              hipStream_t stream) {
  const float* x      = (const float*)d_in[0];
  const float* W_in   = (const float*)d_in[1];
  const float* b_in   = (const float*)d_in[2];
  const float* W_conv = (const float*)d_in[3];
  const float* b_conv = (const float*)d_in[4];
  const float* W_xp   = (const float*)d_in[5];
  const float* b_xp   = (const float*)d_in[6];
  const float* W_dt   = (const float*)d_in[7];
  const float* b_dt   = (const float*)d_in[8];
  const float* W_out  = (const float*)d_in[9];
  const float* b_out  = (const float*)d_in[10];
  const float* A_log  = (const float*)d_in[11];
  const float* D_par  = (const float*)d_in[12];
  float* out = (float*)d_out;

  // Workspace layout (96 MB total; every region fully rewritten per call):
  //  [0*SZ) xi        (reused as dt after conv consumes xi)
  //  [1*SZ) silu(z)
  //  [2*SZ) ut=xc     (overwritten in place with gated y during scanC)
  //  [3*SZ) dpre      (reused as hloc[8MB] + aprod[8MB] after gemm_dt)
  //  [4*SZ) B_sel
  //  [5*SZ) C_sel
  char* ws = (char*)d_ws;
  const size_t SZ = (size_t)kT * 64 * sizeof(float);        // 16 MB
  float* xi   = (float*)(ws + 0 * SZ);
  float* zs   = (float*)(ws + 1 * SZ);
  float* ut   = (float*)(ws + 2 * SZ);
  float* dpre = (float*)(ws + 3 * SZ);
  float* bt   = (float*)(ws + 4 * SZ);
  float* ct   = (float*)(ws + 5 * SZ);
  float* dt   = xi;
  float2* hloc  = (float2*)dpre;
  float2* aprod = (float2*)(ws + 3 * SZ + SZ / 2);

  k_gemm_in <<<kT / 16,              256, 0, stream>>>(x, W_in, b_in, xi, zs);
  k_conv    <<<(kT * 64) / 256,      256, 0, stream>>>(xi, W_conv, b_conv, ut);
  k_gemm_xp <<<(kT / 16) * 12 / 8,   256, 0, stream>>>(ut, W_xp, b_xp, dpre, bt, ct);
  k_gemm_dt <<<(kT / 16) * 4 / 8,    256, 0, stream>>>(dpre, W_dt, b_dt, dt);
  k_scanA   <<<(kB * 64 * kNChunk) / 8, 256, 0, stream>>>(dt, bt, ut, A_log, hloc, aprod);
  k_scanB   <<<(kB * 64 * 64) / 256, 256, 0, stream>>>((float*)hloc, (const float*)aprod);
  k_scanC   <<<(kB * 64 * kNChunk) / 8, 256, 0, stream>>>(dt, bt, ut, ct, zs, hloc, A_log, D_par);
  k_gemm_out<<<(kT / 16) * 2 / 8,    256, 0, stream>>>(ut, W_out, b_out, out);
}